// S5Layer_80195629351272
// MI455X (gfx1250) — compile-verified
//
#include <hip/hip_runtime.h>
#include <stdint.h>

typedef __attribute__((ext_vector_type(2))) float v2f;
typedef __attribute__((ext_vector_type(8))) float v8f;

#define S_LEN   4096
#define BATCH   16
#define N_IN    64
#define N_H     128
#define T_CHUNK 16
#define N_CHUNKS (S_LEN / T_CHUNK)   // 256
#define STATES   (BATCH * N_H)       // 2048

// workspace layout (float offsets)
#define WS_BBR  0                    // bbar_re [64][128]
#define WS_BBI  8192                 // bbar_im [64][128]
#define WS_LBAR 16384                // lambda_bar [128] float2
#define WS_LAMT 16640                // lambda_bar^T [128] float2
#define WS_LPOW 16896                // lambda_bar^(t+1) [16][128] float2 -> 4096 floats
#define WS_CLOC 20992                // carry_local [256][2048] float2 -> 1048576 floats
#define WS_CEXC 1069568              // carry_excl  [256][2048] float2 -> 1048576 floats
// total = 2,118,144 floats (~8.5 MB)

#define K4_LDS_BYTES (131072 + 65536)   // c_mat (32768 f) + Xre/Xim (8192 f each)

__device__ __forceinline__ v8f vzero() {
    v8f v = {0.f, 0.f, 0.f, 0.f, 0.f, 0.f, 0.f, 0.f};
    return v;
}

__device__ __forceinline__ v8f cwmma(v2f a, v2f b, v8f c) {
    // D = A(16x4) * B(4x16) + C(16x16), f32
    return __builtin_amdgcn_wmma_f32_16x16x4_f32(false, a, false, b, (short)0, c, false, false);
}

// ---------------- K1: discretization + tables ----------------
__global__ void k_setup(const float2* __restrict__ lam,
                        const float2* __restrict__ bmat,
                        const float*  __restrict__ delta_p,
                        float* __restrict__ ws) {
    int h = threadIdx.x;               // 128 threads
    float dt = delta_p[0];
    float2 l = lam[h];
    float er  = expf(l.x * dt);
    float lbr = er * cosf(l.y * dt);
    float lbi = er * sinf(l.y * dt);
    // coef = (lambda_bar - 1) / lambda
    float nr = lbr - 1.0f, ni = lbi;
    float den = l.x * l.x + l.y * l.y;
    float cr = (nr * l.x + ni * l.y) / den;
    float ci = (ni * l.x - nr * l.y) / den;
    float* bbr = ws + WS_BBR;
    float* bbi = ws + WS_BBI;
    for (int i = 0; i < N_IN; ++i) {   // store b_bar^T as [k=i][n=h] planes
        float2 b = bmat[h * N_IN + i];
        bbr[i * N_H + h] = cr * b.x - ci * b.y;
        bbi[i * N_H + h] = cr * b.y + ci * b.x;
    }
    ((float2*)(ws + WS_LBAR))[h] = make_float2(lbr, lbi);
    // lambda_bar^16 via 4 squarings
    float pr = lbr, pi = lbi;
    for (int q = 0; q < 4; ++q) { float t = pr * pr - pi * pi; pi = 2.f * pr * pi; pr = t; }
    ((float2*)(ws + WS_LAMT))[h] = make_float2(pr, pi);
    // lambda_bar^(t+1), t = 0..15
    float2* lpow = (float2*)(ws + WS_LPOW);
    float qr = lbr, qi = lbi;
    for (int t = 0; t < T_CHUNK; ++t) {
        lpow[t * N_H + h] = make_float2(qr, qi);
        float n2 = qr * lbr - qi * lbi;
        qi = qr * lbi + qi * lbr;
        qr = n2;
    }
}

// ---------------- K2: fused input GEMM + local chunk scan ----------------
// grid: 256 blocks (one chunk each), 256 threads (8 waves); wave w owns H-tile n0=w*16.
__global__ void __launch_bounds__(256) k_scan_chunk(const float4* __restrict__ u4,
                                                    float* __restrict__ ws,
                                                    float* __restrict__ xs /* = d_out temp */) {
    const int chunk = blockIdx.x;
    const int lane  = threadIdx.x & 31;
    const int wv    = threadIdx.x >> 5;
    const int n0    = wv * 16;
    const int hi    = lane >> 4;   // 0/1: which K pair / which M half in C
    const int ml    = lane & 15;
    const int h     = n0 + ml;

    const float* bbr = ws + WS_BBR;
    const float* bbi = ws + WS_BBI;
    const float2 lb  = ((const float2*)(ws + WS_LBAR))[h];

    // Preload all B fragments of b_bar^T for this wave's n-tile (K=64 -> 16 slices of 4)
    v2f br[16], bi[16];
#pragma unroll
    for (int kk = 0; kk < 16; ++kk) {
        int kb = kk * 4 + hi * 2;
        br[kk].x = bbr[kb * N_H + h];       br[kk].y = bbr[(kb + 1) * N_H + h];
        bi[kk].x = bbi[kb * N_H + h];       bi[kk].y = bbi[(kb + 1) * N_H + h];
    }

    v8f xr = vzero(), xi = vzero();        // scan state, lives in C-matrix layout

    for (int t = 0; t < T_CHUNK; ++t) {
        const int s = chunk * T_CHUNK + t;
        v8f arr = vzero(), aii = vzero(), ari = vzero(), air = vzero();
        const int rowbase = (s * BATCH + ml) * 32;   // float4 index base, row m=ml
#pragma unroll
        for (int kk = 0; kk < 16; ++kk) {
            float4 av = u4[rowbase + kk * 2 + hi];   // (re0, im0, re1, im1)
            v2f ar2; ar2.x = av.x; ar2.y = av.z;
            v2f ai2; ai2.x = av.y; ai2.y = av.w;
            arr = cwmma(ar2, br[kk], arr);
            aii = cwmma(ai2, bi[kk], aii);
            ari = cwmma(ar2, bi[kk], ari);
            air = cwmma(ai2, br[kk], air);
        }
        v8f vr = arr - aii;
        v8f vi = ari + air;
#pragma unroll
        for (int j = 0; j < 8; ++j) {      // x = lambda_bar * x + v (lambda depends on h only)
            float nxr = lb.x * xr[j] - lb.y * xi[j] + vr[j];
            float nxi = lb.x * xi[j] + lb.y * xr[j] + vi[j];
            xr[j] = nxr; xi[j] = nxi;
        }
#pragma unroll
        for (int j = 0; j < 8; ++j) {      // write local xs
            int b = j + 8 * hi;
            ((float2*)xs)[(s * BATCH + b) * N_H + h] = make_float2(xr[j], xi[j]);
        }
    }
    // chunk carry (local scan's last value)
    float2* cl = (float2*)(ws + WS_CLOC);
#pragma unroll
    for (int j = 0; j < 8; ++j) {
        int b = j + 8 * hi;
        cl[chunk * STATES + b * N_H + h] = make_float2(xr[j], xi[j]);
    }
}

// ---------------- K3: exclusive scan of chunk carries ----------------
// grid: 8 blocks x 256 threads -> one (b,h) state per thread; coalesced per-chunk I/O.
__global__ void k_carry_scan(const float2* __restrict__ x0, float* __restrict__ ws) {
    const int idx = blockIdx.x * 256 + threadIdx.x;   // 0..2047
    const float2 lt = ((const float2*)(ws + WS_LAMT))[idx & (N_H - 1)];
    const float2* cl = (const float2*)(ws + WS_CLOC);
    float2* ce = (float2*)(ws + WS_CEXC);
    float2 cur = x0[idx];
#pragma unroll 8
    for (int c = 0; c < N_CHUNKS; ++c) {
        ce[c * STATES + idx] = cur;
        float2 lo = cl[c * STATES + idx];
        float nr2 = lt.x * cur.x - lt.y * cur.y + lo.x;
        float ni2 = lt.x * cur.y + lt.y * cur.x + lo.y;
        cur = make_float2(nr2, ni2);
    }
}

// ---------------- K4: carry fix-up + output GEMM (in place over d_out) ----------------
// grid: 1024 blocks, 64 rows each (rows/chunk = 256, so a 64-row block lies in one chunk).
// c_mat is staged into LDS with async copies overlapped with the fix-up pass.
__global__ void __launch_bounds__(256) k_fix_gemm(const float* __restrict__ cmat,
                                                  float* __restrict__ ws,
                                                  float* __restrict__ out) {
    extern __shared__ float lds[];
    float* Cm  = lds;            // 32768 floats: c_mat verbatim (interleaved complex)
    float* Xre = lds + 32768;    // [64][128]
    float* Xim = lds + 40960;    // [64][128]

    const int r0 = blockIdx.x * 64;
    const int chunk = r0 >> 8;         // 256 rows per chunk
    const float2* lpow = (const float2*)(ws + WS_LPOW);
    const float2* ce   = (const float2*)(ws + WS_CEXC);

    // Kick off async staging of c_mat (131072 B) into LDS: 8192 x b128, 32 per thread.
    {
        uint32_t ldsbase = (uint32_t)(uintptr_t)Cm;   // low 32 bits of LDS pointer = LDS offset
#pragma unroll
        for (int i = 0; i < 32; ++i) {
            uint32_t off = (uint32_t)threadIdx.x * 16u + (uint32_t)i * 4096u;
            uint32_t ldsaddr = ldsbase + off;
            asm volatile("global_load_async_to_lds_b128 %0, %1, %2"
                         :: "v"(ldsaddr), "v"(off), "s"(cmat)
                         : "memory");
        }
    }

    // phase 1 (overlapped with async copy): load xs tile, apply lambda^(t+1)*carry, stage to LDS
    for (int e = threadIdx.x; e < 64 * N_H; e += 256) {
        int col = e & (N_H - 1), rowl = e >> 7;
        int r = r0 + rowl;
        int s = r >> 4, b = r & 15;
        int t = s & (T_CHUNK - 1);
        float2 x  = ((const float2*)out)[r * N_H + col];
        float2 c  = ce[chunk * STATES + b * N_H + col];
        float2 lp = lpow[t * N_H + col];
        Xre[rowl * N_H + col] = x.x + lp.x * c.x - lp.y * c.y;
        Xim[rowl * N_H + col] = x.y + lp.x * c.y + lp.y * c.x;
    }
    asm volatile("s_wait_asynccnt 0" ::: "memory");
    __syncthreads();

    // phase 2: Y = X @ c_mat^T  (B[k][n] = c_mat[n*H + k]), B-frags from LDS
    const int lane = threadIdx.x & 31, wv = threadIdx.x >> 5;
    const int n0 = wv * 16, hi = lane >> 4, ml = lane & 15;
    const int n = n0 + ml;
    for (int mt = 0; mt < 4; ++mt) {
        v8f arr = vzero(), aii = vzero(), ari = vzero(), air = vzero();
#pragma unroll 4
        for (int kk = 0; kk < 32; ++kk) {
            int kb = kk * 4 + hi * 2;
            int m = mt * 16 + ml;
            v2f ar2; ar2.x = Xre[m * N_H + kb]; ar2.y = Xre[m * N_H + kb + 1];
            v2f ai2; ai2.x = Xim[m * N_H + kb]; ai2.y = Xim[m * N_H + kb + 1];
            const float4 bv = *(const float4*)&Cm[(n * N_H + kb) * 2];
            v2f br2; br2.x = bv.x; br2.y = bv.z;
            v2f bi2; bi2.x = bv.y; bi2.y = bv.w;
            arr = cwmma(ar2, br2, arr);
            aii = cwmma(ai2, bi2, aii);
            ari = cwmma(ar2, bi2, ari);
            air = cwmma(ai2, br2, air);
        }
        v8f yr = arr - aii;
        v8f yi = ari + air;
#pragma unroll
        for (int j = 0; j < 8; ++j) {
            int m = mt * 16 + j + 8 * hi;
            ((float2*)out)[(r0 + m) * N_H + n] = make_float2(yr[j], yi[j]);
        }
    }
}

extern "C" void kernel_launch(void* const* d_in, const int* in_sizes, int n_in,
                              void* d_out, int out_size, void* d_ws, size_t ws_size,
                              hipStream_t stream) {
    (void)in_sizes; (void)n_in; (void)out_size; (void)ws_size;
    const float*  u      = (const float*)d_in[0];   // [S,B,IN] complex interleaved
    const float2* lam    = (const float2*)d_in[1];  // [H] complex
    const float2* bmat   = (const float2*)d_in[2];  // [H,IN] complex
    const float*  cmat   = (const float*)d_in[3];   // [H,H] complex interleaved
    const float2* x0     = (const float2*)d_in[4];  // [B,H] complex
    const float*  delta  = (const float*)d_in[5];   // scalar
    float* out = (float*)d_out;
    float* ws  = (float*)d_ws;

    // allow 192 KB dynamic LDS for K4 (CDNA5 WGP: up to 320 KB per workgroup)
    (void)hipFuncSetAttribute(reinterpret_cast<const void*>(k_fix_gemm),
                              hipFuncAttributeMaxDynamicSharedMemorySize, K4_LDS_BYTES);

    k_setup<<<1, N_H, 0, stream>>>(lam, bmat, delta, ws);
    k_scan_chunk<<<N_CHUNKS, 256, 0, stream>>>((const float4*)u, ws, out);
    k_carry_scan<<<STATES / 256, 256, 0, stream>>>(x0, ws);
    k_fix_gemm<<<S_LEN * BATCH / 64, 256, K4_LDS_BYTES, stream>>>(cmat, ws, out);
}